// Model_33818572489041
// MI455X (gfx1250) — compile-verified
//
#include <hip/hip_runtime.h>
#include <hip/hip_bf16.h>
#include <math.h>

// ---------------------------------------------------------------------------
// fp32 WMMA (V_WMMA_F32_16X16X4_F32) helpers
// ---------------------------------------------------------------------------
typedef __attribute__((ext_vector_type(2))) float v2f;
typedef __attribute__((ext_vector_type(8))) float v8f;

static __device__ __forceinline__ v8f wmma_f32_k4(v2f a, v2f b, v8f c) {
    // 8 args: (neg_a, A, neg_b, B, c_mod, C, reuse_a, reuse_b)
    return __builtin_amdgcn_wmma_f32_16x16x4_f32(false, a, false, b, (short)0, c,
                                                 false, false);
}

// Compute a 16x16 tile of  D = A(16xK, LDS row-major lda) * W^T  where W is
// (N,K) row-major in global memory and the tile covers columns [n0, n0+16).
static __device__ __forceinline__ v8f gemm_tile_f32(const float* __restrict__ lds_a,
                                                    int lda,
                                                    const float* __restrict__ w,
                                                    int ldw, int n0, int K,
                                                    int lane) {
    const int mrow = lane & 15;
    const int hi   = lane >> 4;
    const float* arow = lds_a + mrow * lda;
    const float* brow = w + (size_t)(n0 + mrow) * (size_t)ldw;
    v8f acc = {};
    for (int k0 = 0; k0 < K; k0 += 4) {
        const int k = k0 + 2 * hi;
        v2f a; a.x = arow[k]; a.y = arow[k + 1];
        v2f b; b.x = brow[k]; b.y = brow[k + 1];
        acc = wmma_f32_k4(a, b, acc);
    }
    return acc;
}

static __device__ __forceinline__ float sigmoidf_(float x) {
    return 1.0f / (1.0f + __expf(-x));
}

// ---------------------------------------------------------------------------
// Fused conv stack: x[4,10,10] -> relu conv3x3 -> [128,8,8] -> relu conv2x2
// -> [64,7,7] -> relu conv2x2 -> img[64,6,6] = 2304 floats per element.
// One block per batch element; activations staged in LDS.
// ---------------------------------------------------------------------------
#define CONV_LDS_FLOATS (400 + 4608 + 8192 + 3136)

__global__ __launch_bounds__(256) void conv_fused_kernel(
    const float* __restrict__ x, const float* __restrict__ w1,
    const float* __restrict__ b1, const float* __restrict__ w2,
    const float* __restrict__ b2, const float* __restrict__ w3,
    const float* __restrict__ b3, float* __restrict__ g_img) {
    extern __shared__ float sm[];
    float* xs  = sm;            // 4*10*10   = 400
    float* ws1 = xs + 400;      // 128*4*3*3 = 4608
    float* o1  = ws1 + 4608;    // 128*8*8   = 8192
    float* o2  = o1 + 8192;     // 64*7*7    = 3136

    const int tid = threadIdx.x;
    const int bidx = blockIdx.x;
    const float* xin = x + (size_t)bidx * 400;

    for (int e = tid; e < 400; e += 256) xs[e] = xin[e];
    for (int e = tid; e < 4608; e += 256) ws1[e] = w1[e];
    __syncthreads();

    // conv1: 128 x 8 x 8, k=3
    for (int v = tid; v < 8192; v += 256) {
        const int co = v >> 6;
        const int oy = (v >> 3) & 7;
        const int ox = v & 7;
        float s = b1[co];
        const float* wr = ws1 + co * 36;
        for (int ci = 0; ci < 4; ++ci) {
            const float* xp = xs + ci * 100 + oy * 10 + ox;
            const float* wp = wr + ci * 9;
            for (int ky = 0; ky < 3; ++ky)
                for (int kx = 0; kx < 3; ++kx)
                    s += xp[ky * 10 + kx] * wp[ky * 3 + kx];
        }
        o1[v] = s > 0.0f ? s : 0.0f;
    }
    __syncthreads();

    // conv2: 64 x 7 x 7, k=2, Cin=128  (weights from global / L2)
    for (int v = tid; v < 3136; v += 256) {
        const int co = v / 49;
        const int p  = v - co * 49;
        const int oy = p / 7;
        const int ox = p - oy * 7;
        float s = b2[co];
        for (int ci = 0; ci < 128; ++ci) {
            const float* ip = o1 + ci * 64 + oy * 8 + ox;
            const float* wp = w2 + ((co * 128 + ci) << 2);
            s += ip[0] * wp[0] + ip[1] * wp[1] + ip[8] * wp[2] + ip[9] * wp[3];
        }
        o2[v] = s > 0.0f ? s : 0.0f;
    }
    __syncthreads();

    // conv3: 64 x 6 x 6, k=2, Cin=64 -> write to global img
    float* imgo = g_img + (size_t)bidx * 2304;
    for (int v = tid; v < 2304; v += 256) {
        const int co = v / 36;
        const int p  = v - co * 36;
        const int oy = p / 6;
        const int ox = p - oy * 6;
        float s = b3[co];
        for (int ci = 0; ci < 64; ++ci) {
            const float* ip = o2 + ci * 49 + oy * 7 + ox;
            const float* wp = w3 + ((co * 64 + ci) << 2);
            s += ip[0] * wp[0] + ip[1] * wp[1] + ip[7] * wp[2] + ip[8] * wp[3];
        }
        imgo[v] = s > 0.0f ? s : 0.0f;
    }
}

// ---------------------------------------------------------------------------
// Fused GRU(16 steps) + head, one block per 16 batch rows (256 threads).
// The GRU recurrence is block-local (GEMMs only mix across the hidden dim),
// so h stays resident in LDS for all 16 steps. Afterwards the same block
// computes attn, FiLM gating + on-the-fly positional encoding, the 2304->256
// linear (WMMA, K=2304) and the critic, writing out[16,4].
//
// LDS arena (floats):
//   sh_h : 16*256 = 4096                       (persistent)
//   GRU phase  : emb[16 steps][16][32] = 8192 | gi 16*768 | gh 16*768 = 32768
//   head phase : attn 1024 | pos 32 | flat 16*2304 = 36864 | hid 4096 = 42016
// total = 4096 + max(32768, 42016) = 46112 floats = 184448 B  (< 320 KB/WGP)
// ---------------------------------------------------------------------------
#define FUSED_LDS_FLOATS (4096 + 42016)

__global__ __launch_bounds__(256) void gru_head_fused_kernel(
    const int* __restrict__ inst, const float* __restrict__ emb_w,
    const float* __restrict__ w_ih, const float* __restrict__ w_hh,
    const float* __restrict__ b_ih, const float* __restrict__ b_hh,
    const float* __restrict__ g_img, const int* __restrict__ steps,
    const float* __restrict__ attn_w, const float* __restrict__ attn_b,
    const float* __restrict__ lin_w, const float* __restrict__ lin_b,
    const float* __restrict__ crit_w, const float* __restrict__ crit_b,
    float* __restrict__ out) {
    extern __shared__ float sm[];
    float* sh_h   = sm;           // 4096, persistent
    float* arena  = sm + 4096;
    // GRU-phase aliases
    float* sh_e   = arena;                  // 8192   (emb, all 16 steps)
    float* sh_gi  = arena + 8192;           // 12288
    float* sh_gh  = arena + 8192 + 12288;   // 12288
    // head-phase aliases (reuse arena after GRU completes)
    float* sh_attn = arena;                 // 1024
    float* sh_pos  = arena + 1024;          // 32
    float* sh_flat = arena + 1056;          // 36864
    float* sh_hid  = arena + 1056 + 36864;  // 4096

    const int tid = threadIdx.x;
    const int m0  = blockIdx.x * 16;
    const int lane = tid & 31;
    const int wv   = tid >> 5;
    const int hi   = lane >> 4;
    const int mrow = lane & 15;

    // h0 = 0; gather embeddings for all 16 timesteps up front.
    for (int e = tid; e < 16 * 256; e += 256) sh_h[e] = 0.0f;
    for (int e = tid; e < 16 * 16 * 32; e += 256) {
        const int t = e >> 9;          // step
        const int m = (e >> 5) & 15;   // row in tile
        const int d = e & 31;          // embed dim
        const int idx = inst[(m0 + m) * 16 + t];
        sh_e[e] = emb_w[idx * 32 + d];
    }
    __syncthreads();

    // ---------------- GRU: 16 steps, h resident in LDS -----------------
    for (int t = 0; t < 16; ++t) {
        const float* emb_t = sh_e + t * 512;  // [16][32]
        // 48 N-tiles of 16 over the 768 gate dim; 6 per wave.
        for (int i = 0; i < 6; ++i) {
            const int n0 = (wv * 6 + i) * 16;
            if (i < 5)  // prefetch next tile's w_hh rows (global_prefetch_b8)
                __builtin_prefetch(w_hh + (size_t)(n0 + 16 + mrow) * 256, 0, 1);
            v8f agh = gemm_tile_f32(sh_h, 256, w_hh, 256, n0, 256, lane);
            for (int r = 0; r < 8; ++r)
                sh_gh[(r + 8 * hi) * 768 + n0 + mrow] = agh[r];
            v8f agi = gemm_tile_f32(emb_t, 32, w_ih, 32, n0, 32, lane);
            for (int r = 0; r < 8; ++r)
                sh_gi[(r + 8 * hi) * 768 + n0 + mrow] = agi[r];
        }
        __syncthreads();

        // Gates: 16 rows x 256 cols, 16 per thread; h updated in place.
        const int m  = tid >> 4;
        const int c0 = (tid & 15) * 16;
        for (int cc = 0; cc < 16; ++cc) {
            const int c = c0 + cc;
            const float i_r = sh_gi[m * 768 + c]       + b_ih[c];
            const float i_z = sh_gi[m * 768 + 256 + c] + b_ih[256 + c];
            const float i_n = sh_gi[m * 768 + 512 + c] + b_ih[512 + c];
            const float h_r = sh_gh[m * 768 + c]       + b_hh[c];
            const float h_z = sh_gh[m * 768 + 256 + c] + b_hh[256 + c];
            const float h_n = sh_gh[m * 768 + 512 + c] + b_hh[512 + c];
            const float r = sigmoidf_(i_r + h_r);
            const float z = sigmoidf_(i_z + h_z);
            const float n = tanhf(i_n + r * h_n);
            const float hp = sh_h[m * 256 + c];
            sh_h[m * 256 + c] = (1.0f - z) * n + z * hp;
        }
        __syncthreads();
    }

    // ---------------- head ----------------
    if (tid < 16) sh_pos[tid] = (float)steps[m0 + tid];

    // attn: 4 N-tiles over 64 cols; waves 0..3 (wave-uniform branch)
    if (wv < 4) {
        const int n0 = wv * 16;
        v8f acc = gemm_tile_f32(sh_h, 256, attn_w, 256, n0, 256, lane);
        for (int r = 0; r < 8; ++r) {
            const int row = r + 8 * hi;
            const int col = n0 + mrow;
            sh_attn[row * 64 + col] = sigmoidf_(acc[r] + attn_b[col]);
        }
    }
    __syncthreads();

    // flat = img * attn[channel] + PE[pos]  (PE recomputed on device)
    const float kFreq = -9.210340371976184f / 2304.0f;  // -ln(10000)/D_PE
    for (int e = tid; e < 16 * 2304; e += 256) {
        const int m = e / 2304;
        const int j = e - m * 2304;
        const int ch = j / 36;  // 6*6 spatial per channel
        const float freq = __expf((float)(j & ~1) * kFreq);
        const float ang  = sh_pos[m] * freq;
        const float pe   = (j & 1) ? __cosf(ang) : __sinf(ang);
        sh_flat[e] = g_img[(size_t)(m0 + m) * 2304 + j] * sh_attn[m * 64 + ch] + pe;
    }
    __syncthreads();

    // lin: 16 N-tiles over 256 cols, K = 2304; 2 tiles per wave
    for (int i = 0; i < 2; ++i) {
        const int n0 = (wv * 2 + i) * 16;
        v8f acc = gemm_tile_f32(sh_flat, 2304, lin_w, 2304, n0, 2304, lane);
        for (int r = 0; r < 8; ++r) {
            const int row = r + 8 * hi;
            const int col = n0 + mrow;
            const float v = acc[r] + lin_b[col];
            sh_hid[row * 256 + col] = v > 0.0f ? v : 0.0f;
        }
    }
    __syncthreads();

    // critic: [16,256] @ [256,4]^T
    if (tid < 64) {
        const int m = tid >> 2;
        const int o = tid & 3;
        float s = crit_b[o];
        const float* hr = sh_hid + m * 256;
        const float* wr = crit_w + o * 256;
        for (int k = 0; k < 256; ++k) s += hr[k] * wr[k];
        out[(size_t)(m0 + m) * 4 + o] = s;
    }
}

// ---------------------------------------------------------------------------
// Launcher: two kernels total (conv stack, then fused GRU+head).
// ---------------------------------------------------------------------------
extern "C" void kernel_launch(void* const* d_in, const int* in_sizes, int n_in,
                              void* d_out, int out_size, void* d_ws, size_t ws_size,
                              hipStream_t stream) {
    const float* x      = (const float*)d_in[0];   // [B,4,10,10]
    const int*   inst   = (const int*)d_in[1];     // [B,16]
    const int*   steps  = (const int*)d_in[2];     // [B,1]
    const float* img1_w = (const float*)d_in[3];
    const float* img1_b = (const float*)d_in[4];
    const float* img2_w = (const float*)d_in[5];
    const float* img2_b = (const float*)d_in[6];
    const float* img3_w = (const float*)d_in[7];
    const float* img3_b = (const float*)d_in[8];
    const float* emb_w  = (const float*)d_in[9];
    const float* w_ih   = (const float*)d_in[10];
    const float* w_hh   = (const float*)d_in[11];
    const float* b_ih   = (const float*)d_in[12];
    const float* b_hh   = (const float*)d_in[13];
    const float* attn_w = (const float*)d_in[14];
    const float* attn_b = (const float*)d_in[15];
    const float* lin_w  = (const float*)d_in[16];
    const float* lin_b  = (const float*)d_in[17];
    const float* crit_w = (const float*)d_in[18];
    const float* crit_b = (const float*)d_in[19];
    float* out = (float*)d_out;

    const int B = in_sizes[0] / 400;

    // Workspace: img [B,2304] only (h lives in LDS now).
    float* g_img = (float*)d_ws;

    conv_fused_kernel<<<B, 256, CONV_LDS_FLOATS * sizeof(float), stream>>>(
        x, img1_w, img1_b, img2_w, img2_b, img3_w, img3_b, g_img);

    gru_head_fused_kernel<<<B / 16, 256, FUSED_LDS_FLOATS * sizeof(float), stream>>>(
        inst, emb_w, w_ih, w_hh, b_ih, b_hh, g_img, steps, attn_w, attn_b,
        lin_w, lin_b, crit_w, crit_b, out);
}